// SelectiveSSM_53824530153925
// MI455X (gfx1250) — compile-verified
//
#include <hip/hip_runtime.h>
#include <hip/hip_bf16.h>
#include <stdint.h>

// Problem constants (match reference)
#define BSZ   2
#define LSEQ  2048
#define DIN   2048
#define NST   16
#define RNK   128
#define EOUT  160              // DT_RANK + 2*N
#define MROW  (BSZ * LSEQ)     // 4096

typedef __attribute__((ext_vector_type(2))) float v2f;
typedef __attribute__((ext_vector_type(8))) float v8f;

// ---------------------------------------------------------------------------
// Kernel 1: xp[m][e] = sum_k x[m][k] * Wx[e][k]   (M=4096, N=160, K=2048)
// One wave per 16x16 output tile, V_WMMA_F32_16X16X4_F32 over K.
// ---------------------------------------------------------------------------
__global__ __launch_bounds__(320)
void ssm_gemm_xp(const float* __restrict__ x, const float* __restrict__ Wx,
                 float* __restrict__ xp) {
    const int lane = threadIdx.x & 31;
    const int wv   = threadIdx.x >> 5;       // 0..9 -> e tile
    const int half = lane >> 4;              // 0: K lanes 0/1, 1: K lanes 2/3
    const int l16  = lane & 15;

    const int m0 = blockIdx.x * 16;
    const int e0 = wv * 16;

    const float* arow = x  + (size_t)(m0 + l16) * DIN + half * 2;
    const float* brow = Wx + (size_t)(e0 + l16) * DIN + half * 2;

    v8f c = {};
#pragma unroll 4
    for (int k0 = 0; k0 < DIN; k0 += 4) {
        v2f a = *(const v2f*)(arow + k0);    // A: row m0+l16, K = k0+half*2, +1
        v2f b = *(const v2f*)(brow + k0);    // B: col e0+l16, K = k0+half*2, +1
        c = __builtin_amdgcn_wmma_f32_16x16x4_f32(
                false, a, false, b, (short)0, c, false, false);
    }

#pragma unroll
    for (int r = 0; r < 8; ++r) {
        const int m = m0 + r + half * 8;     // C/D layout: vgpr r, lane half
        xp[(size_t)m * EOUT + e0 + l16] = c[r];
    }
}

// ---------------------------------------------------------------------------
// Kernel 2: delta[m][d] = softplus( sum_r xp[m][r]*Wdt[d][r] + bdt[d] )
// M=4096, N=2048, K=128.  Grid = (16, 256); 8 waves/block, one N-tile/wave.
// ---------------------------------------------------------------------------
__global__ __launch_bounds__(256)
void ssm_gemm_delta(const float* __restrict__ xp, const float* __restrict__ Wdt,
                    const float* __restrict__ bdt, float* __restrict__ delta) {
    const int lane = threadIdx.x & 31;
    const int wv   = threadIdx.x >> 5;       // 0..7
    const int half = lane >> 4;
    const int l16  = lane & 15;

    const int m0 = blockIdx.y * 16;
    const int d0 = (blockIdx.x * 8 + wv) * 16;

    const float* arow = xp  + (size_t)(m0 + l16) * EOUT + half * 2;
    const float* brow = Wdt + (size_t)(d0 + l16) * RNK  + half * 2;

    v8f c = {};
#pragma unroll
    for (int k0 = 0; k0 < RNK; k0 += 4) {
        v2f a = *(const v2f*)(arow + k0);
        v2f b = *(const v2f*)(brow + k0);
        c = __builtin_amdgcn_wmma_f32_16x16x4_f32(
                false, a, false, b, (short)0, c, false, false);
    }

    const int   d  = d0 + l16;
    const float bb = bdt[d];
#pragma unroll
    for (int r = 0; r < 8; ++r) {
        const int m = m0 + r + half * 8;
        float v  = c[r] + bb;
        float sp = (v > 20.0f) ? v : __logf(1.0f + __expf(v));  // softplus
        delta[(size_t)m * DIN + d] = sp;
    }
}

// ---------------------------------------------------------------------------
// Kernel 3: fused selective scan with DOUBLE-BUFFERED ASYNC global->LDS
// staging (GLOBAL_LOAD_ASYNC_TO_LDS_B32, ASYNCcnt) so HBM latency for chunk
// c+1 hides behind the serial exp/FMA chain of chunk c.
//   h[d][n] = exp(delta*A[d][n]) * h + (delta*x[d]) * B[n]
//   y[d]    = sum_n h[d][n] * C[n]  + x[d]*Dp[d]
// One block = (b, 16 channels). 16-lane group = one channel d, lane%16 = n.
// ---------------------------------------------------------------------------
#define TCHUNK  32
#define NCHUNK  (LSEQ / TCHUNK)

// arrays: 0 = delta, 1 = x, 2 = B_mat, 3 = C_mat
__global__ __launch_bounds__(256)
void ssm_scan(const float* __restrict__ x, const float* __restrict__ xp,
              const float* __restrict__ delta, const float* __restrict__ A_log,
              const float* __restrict__ Dp, float* __restrict__ y) {
    __shared__ float sh[2][4][TCHUNK][16];   // 16 KB double buffer

    const int bb = blockIdx.x;               // 0..255
    const int b  = bb >> 7;                  // batch
    const int d0 = (bb & 127) * 16;          // first channel of this block

    const int tid = threadIdx.x;
    const int ch  = tid >> 4;                // 0..15 channel within block
    const int n   = tid & 15;                // state index
    const int d   = d0 + ch;

    const float A_dn = -__expf(A_log[(size_t)d * NST + n]);
    const float dp_d = Dp[d];

    const size_t rowD = (size_t)(b * LSEQ) * DIN  + d0;   // delta / x base
    const size_t rowE = (size_t)(b * LSEQ) * EOUT + RNK;  // B_mat base in xp

    // Issue 8 async b32 loads (2 per array) for chunk starting at t0 into buf.
    auto stage = [&](int buf, int t0) {
#pragma unroll
        for (int k = 0; k < 2; ++k) {
            const int i  = tid + (k << 8);
            const int tt = i >> 4;
            const int dd = i & 15;
            const float* gd = delta + rowD + (size_t)(t0 + tt) * DIN + dd;
            const float* gx = x     + rowD + (size_t)(t0 + tt) * DIN + dd;
            const float* gB = xp    + rowE + (size_t)(t0 + tt) * EOUT + dd;
            const float* gC = gB + NST;
            asm volatile("global_load_async_to_lds_b32 %0, %1, off"
                         :: "v"((unsigned)(uintptr_t)&sh[buf][0][tt][dd]),
                            "v"(gd) : "memory");
            asm volatile("global_load_async_to_lds_b32 %0, %1, off"
                         :: "v"((unsigned)(uintptr_t)&sh[buf][1][tt][dd]),
                            "v"(gx) : "memory");
            asm volatile("global_load_async_to_lds_b32 %0, %1, off"
                         :: "v"((unsigned)(uintptr_t)&sh[buf][2][tt][dd]),
                            "v"(gB) : "memory");
            asm volatile("global_load_async_to_lds_b32 %0, %1, off"
                         :: "v"((unsigned)(uintptr_t)&sh[buf][3][tt][dd]),
                            "v"(gC) : "memory");
        }
    };

    float h = 0.0f;

    stage(0, 0);                                       // prime buffer 0
    asm volatile("s_wait_asynccnt 0" ::: "memory");
    __syncthreads();

    for (int c = 0; c < NCHUNK; ++c) {
        const int cur = c & 1;
        if (c + 1 < NCHUNK)
            stage(cur ^ 1, (c + 1) * TCHUNK);          // prefetch next chunk

#pragma unroll 4
        for (int tt = 0; tt < TCHUNK; ++tt) {
            const float sd = sh[cur][0][tt][ch];
            const float sx = sh[cur][1][tt][ch];
            const float sB = sh[cur][2][tt][n];
            const float sC = sh[cur][3][tt][n];

            const float dA = __expf(sd * A_dn);
            h = __builtin_fmaf(dA, h, (sd * sx) * sB);

            float yv = h * sC;
            // reduce over n (16 lanes) within the wave32
            yv += __shfl_xor(yv, 8, 32);
            yv += __shfl_xor(yv, 4, 32);
            yv += __shfl_xor(yv, 2, 32);
            yv += __shfl_xor(yv, 1, 32);

            if (n == 0) {
                const size_t row = (size_t)(b * LSEQ + c * TCHUNK + tt);
                y[row * DIN + d] = yv + sx * dp_d;
            }
        }

        asm volatile("s_wait_asynccnt 0" ::: "memory"); // next buf resident
        __syncthreads();
    }
}

// ---------------------------------------------------------------------------
extern "C" void kernel_launch(void* const* d_in, const int* in_sizes, int n_in,
                              void* d_out, int out_size, void* d_ws, size_t ws_size,
                              hipStream_t stream) {
    (void)in_sizes; (void)n_in; (void)out_size; (void)ws_size;

    const float* x     = (const float*)d_in[0];  // (2,2048,2048)
    const float* Wx    = (const float*)d_in[1];  // (160,2048)
    const float* Wdt   = (const float*)d_in[2];  // (2048,128)
    const float* bdt   = (const float*)d_in[3];  // (2048,)
    const float* A_log = (const float*)d_in[4];  // (2048,16)
    const float* Dp    = (const float*)d_in[5];  // (2048,)
    float*       y     = (float*)d_out;          // (2,2048,2048)

    // Workspace layout: xp (4096*160 f32) | delta (4096*2048 f32)  ~36.2 MB
    float* xp    = (float*)d_ws;
    float* delta = xp + (size_t)MROW * EOUT;

    // 1) xp = x @ Wx^T
    ssm_gemm_xp<<<dim3(MROW / 16), dim3(320), 0, stream>>>(x, Wx, xp);

    // 2) delta = softplus(xp[:, :128] @ Wdt^T + bdt)
    ssm_gemm_delta<<<dim3(DIN / (8 * 16), MROW / 16), dim3(256), 0, stream>>>(
        xp, Wdt, bdt, delta);

    // 3) fused selective scan + skip connection (async double-buffered)
    ssm_scan<<<dim3(BSZ * DIN / 16), dim3(256), 0, stream>>>(
        x, xp, delta, A_log, Dp, y);
}